// GNN_26242250178821
// MI455X (gfx1250) — compile-verified
//
#include <hip/hip_runtime.h>
#include <math.h>

// ---------------------------------------------------------------------------
// 2-layer GCN (PyG GCNConv semantics) for MI455X / gfx1250.
//   out[L] stacked: [3, N, 64] f32.
//   GEMM done with V_WMMA_F32_16X16X4_F32 (exact f32, wave32 WMMA).
//   Edge gather/scatter is L2-resident (xw = 38.4MB << 192MB L2).
// ---------------------------------------------------------------------------

typedef __attribute__((ext_vector_type(2))) float v2f;
typedef __attribute__((ext_vector_type(8))) float v8f;

#define DIM 64
#define LDS_STRIDE 68   // 64 + 4 pad: conflict-free banks, keeps float2 8B-aligned

// ---------------- degree / normalization ----------------

__global__ void gcn_deg_init(float* __restrict__ deg, int n) {
    int i = blockIdx.x * blockDim.x + threadIdx.x;
    if (i < n) deg[i] = 1.0f;                      // self-loop
}

__global__ void gcn_deg_accum(const int* __restrict__ edge_index,
                              float* __restrict__ deg, int nedges) {
    int e = blockIdx.x * blockDim.x + threadIdx.x;
    if (e < nedges) atomicAdd(&deg[edge_index[nedges + e]], 1.0f);  // col = dst
}

__global__ void gcn_rsqrt_inplace(float* __restrict__ d, int n) {
    int i = blockIdx.x * blockDim.x + threadIdx.x;
    if (i < n) d[i] = rsqrtf(d[i]);
}

// ---------------- layer-0: concat(users, items) ----------------

__global__ void gcn_concat(const float4* __restrict__ users,
                           const float4* __restrict__ items,
                           float4* __restrict__ out0,
                           int nu4, int ntot4) {
    int i = blockIdx.x * blockDim.x + threadIdx.x;
    if (i < ntot4) out0[i] = (i < nu4) ? users[i] : items[i - nu4];
}

// ---------------- WMMA GEMM: xw = (relu?)X @ W ; outL = xw*dinv^2 + b -------
// One wave per 16-row tile; 4 output 16x16 tiles (all 64 cols); K=64 in 16
// steps of V_WMMA_F32_16X16X4_F32. A 16x4 f32 layout: lanes 0-15 hold rows
// M=lane with K=(0,1) in the 2 A VGPRs, lanes 16-31 hold K=(2,3). B mirrored.
// RELU is a compile-time template arg so the relu path is a bare v_max (no
// runtime-flag cndmask chains interleaved with the WMMAs).

template <bool RELU>
__global__ __launch_bounds__(256) void gcn_gemm_wmma(
    const float* __restrict__ X,     // [nrows, 64]  (layer input)
    const float* __restrict__ W,     // [64, 64]
    const float* __restrict__ bias,  // [64]
    const float* __restrict__ dinv,  // [nrows]
    float* __restrict__ xw,          // [nrows, 64]  (workspace)
    float* __restrict__ outL,        // [nrows, 64]  (self-loop + bias init)
    int nrows)
{
    __shared__ float Wt[DIM * LDS_STRIDE];         // W transposed: Wt[n][k]

    int tid = threadIdx.x;
    for (int i = tid; i < DIM * DIM; i += 256) {
        int k = i >> 6, n = i & 63;
        Wt[n * LDS_STRIDE + k] = W[i];
    }
    __syncthreads();

    int wave = tid >> 5;
    int lane = tid & 31;
    long tile = (long)blockIdx.x * 8 + wave;
    if (tile * 16 >= (long)nrows) return;          // nrows is a multiple of 16

    int lo   = lane & 15;                          // M (A) / N (B,C)
    int hi   = lane >> 4;                          // half-wave select
    int koff = hi * 2;                             // K offset: 0 or 2
    long rowA = tile * 16 + lo;
    const float* xr = X + rowA * DIM;

    v8f acc[4];
    acc[0] = v8f{}; acc[1] = v8f{}; acc[2] = v8f{}; acc[3] = v8f{};

#pragma unroll
    for (int ks = 0; ks < 16; ++ks) {
        int kb = ks * 4 + koff;
        v2f a;
        if (RELU) {
            a.x = fmaxf(xr[kb], 0.0f);
            a.y = fmaxf(xr[kb + 1], 0.0f);
        } else {
            a.x = xr[kb];
            a.y = xr[kb + 1];
        }
#pragma unroll
        for (int t = 0; t < 4; ++t) {
            int n = t * 16 + lo;
            v2f b;
            b.x = Wt[n * LDS_STRIDE + kb];
            b.y = Wt[n * LDS_STRIDE + kb + 1];
            acc[t] = __builtin_amdgcn_wmma_f32_16x16x4_f32(
                false, a, false, b, (short)0, acc[t], false, false);
        }
    }

    // C/D layout: VGPR r -> M=r (lanes 0-15), M=8+r (lanes 16-31); N=lane&15.
    long mbase = tile * 16 + hi * 8;
    float d2[8];
#pragma unroll
    for (int r = 0; r < 8; ++r) {
        float di = dinv[mbase + r];
        d2[r] = di * di;
    }
#pragma unroll
    for (int t = 0; t < 4; ++t) {
        int col = t * 16 + lo;
        float bcol = bias[col];
#pragma unroll
        for (int r = 0; r < 8; ++r) {
            long row = mbase + r;
            float v = acc[t][r];
            xw[row * DIM + col]   = v;
            outL[row * DIM + col] = v * d2[r] + bcol;   // self-loop + bias
        }
    }
}

// ---------------- edge message scatter ----------------
// 64 lanes per edge: coalesced 256B row read of xw[row], contiguous 256B of
// f32 atomics into outL[col]. Both arrays are L2-resident (38.4MB < 192MB).

__global__ __launch_bounds__(256) void gcn_scatter(
    const int* __restrict__ edge_index,   // [2, E] flat: rows then cols
    const float* __restrict__ dinv,
    const float* __restrict__ xw,
    float* __restrict__ outL,
    int nedges)
{
    long idx = (long)blockIdx.x * blockDim.x + threadIdx.x;
    long e = idx >> 6;
    int  d = (int)(idx & 63);
    if (e >= (long)nedges) return;
    int r = edge_index[e];
    int c = edge_index[(long)nedges + e];
    float coef = dinv[r] * dinv[c];
    atomicAdd(&outL[(long)c * DIM + d], xw[(long)r * DIM + d] * coef);
}

// ---------------------------------------------------------------------------

extern "C" void kernel_launch(void* const* d_in, const int* in_sizes, int n_in,
                              void* d_out, int out_size, void* d_ws, size_t ws_size,
                              hipStream_t stream) {
    const float* emb_users  = (const float*)d_in[0];
    const float* emb_items  = (const float*)d_in[1];
    const float* W1         = (const float*)d_in[2];
    const float* b1         = (const float*)d_in[3];
    const float* W2         = (const float*)d_in[4];
    const float* b2         = (const float*)d_in[5];
    const int*   edge_index = (const int*)d_in[6];
    // d_in[7] = edge_weight: accepted but unused (matches reference).

    const int nu = in_sizes[0] / DIM;
    const int ni = in_sizes[1] / DIM;
    const int n  = nu + ni;                 // 150000 (multiple of 16)
    const int E  = in_sizes[6] / 2;         // 3,000,000

    float* out  = (float*)d_out;
    float* out0 = out;
    float* out1 = out + (size_t)n * DIM;
    float* out2 = out + 2 * (size_t)n * DIM;

    // workspace: [dinv: n floats (rounded)] [xw: n*64 floats]
    float* dinv = (float*)d_ws;
    size_t dinv_stride = ((size_t)n + 255) & ~(size_t)255;
    float* xw = dinv + dinv_stride;

    const int T = 256;

    // 1) degrees (in-degree at col, +1 self-loop) -> dinv
    gcn_deg_init  <<<(n + T - 1) / T, T, 0, stream>>>(dinv, n);
    gcn_deg_accum <<<(E + T - 1) / T, T, 0, stream>>>(edge_index, dinv, E);
    gcn_rsqrt_inplace<<<(n + T - 1) / T, T, 0, stream>>>(dinv, n);

    // 2) layer 0 output = concat(users, items)
    int ntot4 = n * (DIM / 4);
    int nu4   = nu * (DIM / 4);
    gcn_concat<<<(ntot4 + T - 1) / T, T, 0, stream>>>(
        (const float4*)emb_users, (const float4*)emb_items, (float4*)out0,
        nu4, ntot4);

    int gemm_blocks    = ((n / 16) + 7) / 8;                 // 8 waves/block
    long scatter_items = (long)E * DIM;
    int scatter_blocks = (int)((scatter_items + T - 1) / T);

    // 3) layer 1: xw1 = emb0 @ W1 ; out1 = xw1*dinv^2 + b1 ; scatter messages
    gcn_gemm_wmma<false><<<gemm_blocks, T, 0, stream>>>(out0, W1, b1, dinv,
                                                        xw, out1, n);
    gcn_scatter<<<scatter_blocks, T, 0, stream>>>(edge_index, dinv, xw, out1, E);

    // 4) layer 2: xw2 = relu(out1) @ W2 ; out2 = xw2*dinv^2 + b2 ; scatter
    gcn_gemm_wmma<true><<<gemm_blocks, T, 0, stream>>>(out1, W2, b2, dinv,
                                                       xw, out2, n);
    gcn_scatter<<<scatter_blocks, T, 0, stream>>>(edge_index, dinv, xw, out2, E);
}